// SessionAffineBank_50105088475171
// MI455X (gfx1250) — compile-verified
//
#include <hip/hip_runtime.h>
#include <hip/hip_bf16.h>

// Problem constants (fixed by the reference).
constexpr int Btot = 8192;   // rows
constexpr int Dk   = 1024;   // feature dim
constexpr int Sexp = 32;     // sessions / experts
constexpr int TM   = 64;     // M rows per tile (4 row groups x 16)
constexpr int MAXT = Btot / TM + Sexp;  // worst-case padded tiles = 160
constexpr int NTHR = 512;               // 16 wave32s; N = 16 waves * 64 = 1024 = Dk

// Workspace layout (int indices into d_ws)
constexpr int WS_COUNTS  = 0;     // [32]
constexpr int WS_CURSORS = 32;    // [32]
constexpr int WS_OFFSETS = 64;    // [33]
constexpr int WS_DESC    = 128;   // [3 * MAXT] = (session, startPos, nvalid)
constexpr int WS_ROWIDX  = 2048;  // [Btot]

typedef __attribute__((ext_vector_type(2))) float v2f;
typedef __attribute__((ext_vector_type(8))) float v8f;

// ---------------- Phase 0: zero counters ----------------
__global__ void sab_zero_kernel(int* wsi) {
    int i = blockIdx.x * blockDim.x + threadIdx.x;
    if (i < 128) wsi[i] = 0;  // counts + cursors (+ slack)
}

// ---------------- Phase 1: per-session histogram ----------------
__global__ void sab_count_kernel(const int* __restrict__ sidx, int* __restrict__ wsi) {
    int i = blockIdx.x * blockDim.x + threadIdx.x;
    if (i < Btot) atomicAdd(&wsi[WS_COUNTS + sidx[i]], 1);
}

// ---------------- Phase 2: serial scan + tile descriptors (tiny: S=32) ----------------
__global__ void sab_scan_kernel(int* __restrict__ wsi) {
    if (threadIdx.x != 0 || blockIdx.x != 0) return;
    int off = 0;
    int t = 0;
    for (int s = 0; s < Sexp; ++s) {
        wsi[WS_OFFSETS + s] = off;
        int c = wsi[WS_COUNTS + s];
        int nt = (c + TM - 1) / TM;
        for (int j = 0; j < nt; ++j) {
            int start = off + j * TM;
            int nv = c - j * TM;
            if (nv > TM) nv = TM;
            wsi[WS_DESC + 3 * t + 0] = s;
            wsi[WS_DESC + 3 * t + 1] = start;
            wsi[WS_DESC + 3 * t + 2] = nv;
            ++t;
        }
        off += c;
    }
    wsi[WS_OFFSETS + Sexp] = off;
    for (; t < MAXT; ++t) wsi[WS_DESC + 3 * t + 0] = -1;  // inactive tiles
}

// ---------------- Phase 3: scatter row ids into compacted buckets ----------------
__global__ void sab_scatter_kernel(const int* __restrict__ sidx, int* __restrict__ wsi) {
    int i = blockIdx.x * blockDim.x + threadIdx.x;
    if (i < Btot) {
        int s = sidx[i];
        int pos = wsi[WS_OFFSETS + s] + atomicAdd(&wsi[WS_CURSORS + s], 1);
        wsi[WS_ROWIDX + pos] = i;
    }
}

// ---------------- Phase 4: gathered WMMA GEMM per 64-row session tile ----------------
// Block: 512 threads = 16 wave32s. Tile: M=64 gathered rows (4 groups of 16),
// N=1024 (full D), K=1024. Each wave owns a 64x64 panel = 4x4 WMMA accumulators.
// Every B fragment (W^T, streamed from L2) is reused by FOUR row groups:
// 64 B of W per WMMA instead of 256 B. A fragments from a 256 KB LDS x-stage
// (x is staged exactly once). 16 waves/WGP = 4 per SIMD32.
__global__ __launch_bounds__(NTHR) void sab_gemm_kernel(
    const float* __restrict__ x, const float* __restrict__ W,
    const float* __restrict__ bias, const int* __restrict__ rowidx,
    const int* __restrict__ desc, float* __restrict__ out)
{
    __shared__ float Xs[TM * Dk];   // 256 KB staged gathered rows
    __shared__ int   rows[TM];

    const int tile = blockIdx.x;
    const int sess = desc[3 * tile + 0];
    if (sess < 0) return;                       // uniform: padded tile, no work
    const int start  = desc[3 * tile + 1];
    const int nvalid = desc[3 * tile + 2];

    const int tid = threadIdx.x;
    if (tid < TM) rows[tid] = (tid < nvalid) ? rowidx[start + tid] : -1;
    __syncthreads();

    // Stage TM rows of x into LDS, zeros for padded rows. Coalesced float4 loads.
    for (int i = tid; i < TM * (Dk / 4); i += NTHR) {
        int m  = i >> 8;           // Dk/4 == 256 float4 per row
        int c4 = i & 255;
        int r  = rows[m];
        float4 v = make_float4(0.f, 0.f, 0.f, 0.f);
        if (r >= 0) v = ((const float4*)(x + (size_t)r * Dk))[c4];
        ((float4*)(Xs + m * Dk))[c4] = v;
    }
    __syncthreads();

    const int wave   = tid >> 5;       // 0..15 (wave32)
    const int lane   = tid & 31;
    const int lane16 = lane & 15;      // A: row M   / B: col N
    const int kh     = lane >> 4;      // K half: lanes 16-31 hold K+2..K+3
    const int nBase  = wave * 64;      // 16 waves span full N = 1024

    const float* Wb = W + (size_t)sess * Dk * Dk;
    const float* w0 = Wb + (size_t)(nBase +  0 + lane16) * Dk + 2 * kh;
    const float* w1 = Wb + (size_t)(nBase + 16 + lane16) * Dk + 2 * kh;
    const float* w2 = Wb + (size_t)(nBase + 32 + lane16) * Dk + 2 * kh;
    const float* w3 = Wb + (size_t)(nBase + 48 + lane16) * Dk + 2 * kh;
    const float* xs0 = Xs + (lane16 +  0) * Dk + 2 * kh;   // row group 0
    const float* xs1 = Xs + (lane16 + 16) * Dk + 2 * kh;   // row group 1
    const float* xs2 = Xs + (lane16 + 32) * Dk + 2 * kh;   // row group 2
    const float* xs3 = Xs + (lane16 + 48) * Dk + 2 * kh;   // row group 3

    v8f acc[4][4];
    #pragma unroll
    for (int g = 0; g < 4; ++g)
        #pragma unroll
        for (int t = 0; t < 4; ++t) acc[g][t] = (v8f){};

    #pragma unroll 2
    for (int k0 = 0; k0 < Dk; k0 += 4) {
        v2f a0 = *(const v2f*)(xs0 + k0);   // ds_load_b64
        v2f a1 = *(const v2f*)(xs1 + k0);
        v2f a2 = *(const v2f*)(xs2 + k0);
        v2f a3 = *(const v2f*)(xs3 + k0);
        v2f b0 = *(const v2f*)(w0 + k0);    // global_load_b64 (W row-major -> B = W^T frag)
        v2f b1 = *(const v2f*)(w1 + k0);
        v2f b2 = *(const v2f*)(w2 + k0);
        v2f b3 = *(const v2f*)(w3 + k0);
        acc[0][0] = __builtin_amdgcn_wmma_f32_16x16x4_f32(false, a0, false, b0, (short)0, acc[0][0], false, false);
        acc[1][0] = __builtin_amdgcn_wmma_f32_16x16x4_f32(false, a1, false, b0, (short)0, acc[1][0], false, false);
        acc[2][0] = __builtin_amdgcn_wmma_f32_16x16x4_f32(false, a2, false, b0, (short)0, acc[2][0], false, false);
        acc[3][0] = __builtin_amdgcn_wmma_f32_16x16x4_f32(false, a3, false, b0, (short)0, acc[3][0], false, false);
        acc[0][1] = __builtin_amdgcn_wmma_f32_16x16x4_f32(false, a0, false, b1, (short)0, acc[0][1], false, false);
        acc[1][1] = __builtin_amdgcn_wmma_f32_16x16x4_f32(false, a1, false, b1, (short)0, acc[1][1], false, false);
        acc[2][1] = __builtin_amdgcn_wmma_f32_16x16x4_f32(false, a2, false, b1, (short)0, acc[2][1], false, false);
        acc[3][1] = __builtin_amdgcn_wmma_f32_16x16x4_f32(false, a3, false, b1, (short)0, acc[3][1], false, false);
        acc[0][2] = __builtin_amdgcn_wmma_f32_16x16x4_f32(false, a0, false, b2, (short)0, acc[0][2], false, false);
        acc[1][2] = __builtin_amdgcn_wmma_f32_16x16x4_f32(false, a1, false, b2, (short)0, acc[1][2], false, false);
        acc[2][2] = __builtin_amdgcn_wmma_f32_16x16x4_f32(false, a2, false, b2, (short)0, acc[2][2], false, false);
        acc[3][2] = __builtin_amdgcn_wmma_f32_16x16x4_f32(false, a3, false, b2, (short)0, acc[3][2], false, false);
        acc[0][3] = __builtin_amdgcn_wmma_f32_16x16x4_f32(false, a0, false, b3, (short)0, acc[0][3], false, false);
        acc[1][3] = __builtin_amdgcn_wmma_f32_16x16x4_f32(false, a1, false, b3, (short)0, acc[1][3], false, false);
        acc[2][3] = __builtin_amdgcn_wmma_f32_16x16x4_f32(false, a2, false, b3, (short)0, acc[2][3], false, false);
        acc[3][3] = __builtin_amdgcn_wmma_f32_16x16x4_f32(false, a3, false, b3, (short)0, acc[3][3], false, false);
    }

    // Epilogue: C/D layout = N: lane&15, M: vgpr + 8*(lane>=16) (+16 per row group).
    // Hoist bias (4 values) and row indices (one ds_load per (g,j)) out of the t-loop.
    float bv[4];
    #pragma unroll
    for (int t = 0; t < 4; ++t) bv[t] = bias[sess * Dk + nBase + t * 16 + lane16];

    #pragma unroll
    for (int g = 0; g < 4; ++g) {
        #pragma unroll
        for (int j = 0; j < 8; ++j) {
            int mm = 16 * g + 8 * kh + j;
            int r  = rows[mm];
            if (r >= 0) {
                float* o = out + (size_t)r * Dk + nBase + lane16;
                o[ 0] = acc[g][0][j] + bv[0];
                o[16] = acc[g][1][j] + bv[1];
                o[32] = acc[g][2][j] + bv[2];
                o[48] = acc[g][3][j] + bv[3];
            }
        }
    }
}

extern "C" void kernel_launch(void* const* d_in, const int* in_sizes, int n_in,
                              void* d_out, int out_size, void* d_ws, size_t ws_size,
                              hipStream_t stream) {
    const float* x    = (const float*)d_in[0];
    const float* W    = (const float*)d_in[1];
    const float* bias = (const float*)d_in[2];
    const int*   sidx = (const int*)d_in[3];
    float* out = (float*)d_out;
    int*   wsi = (int*)d_ws;

    sab_zero_kernel<<<1, 128, 0, stream>>>(wsi);
    sab_count_kernel<<<Btot / 256, 256, 0, stream>>>(sidx, wsi);
    sab_scan_kernel<<<1, 32, 0, stream>>>(wsi);
    sab_scatter_kernel<<<Btot / 256, 256, 0, stream>>>(sidx, wsi);

    dim3 grid(MAXT, 1);   // fixed grid -> deterministic, capture-safe
    sab_gemm_kernel<<<grid, NTHR, 0, stream>>>(x, W, bias,
                                               wsi + WS_ROWIDX, wsi + WS_DESC, out);
}